// GraphSAGEEncoder_33681133535935
// MI455X (gfx1250) — compile-verified
//
#include <hip/hip_runtime.h>

typedef __attribute__((ext_vector_type(2))) float v2f;
typedef __attribute__((ext_vector_type(8))) float v8f;

#define DFEAT 128

// ---------------------------------------------------------------------------
// Degree count: one float atomic per edge (counts are identical every layer).
// ---------------------------------------------------------------------------
__global__ void count_deg(const long long* __restrict__ dst,
                          float* __restrict__ cnt, int E) {
    int e = blockIdx.x * blockDim.x + threadIdx.x;
    if (e < E) atomicAdd(cnt + dst[e], 1.0f);
}

// ---------------------------------------------------------------------------
// One-shot conversion cnt -> 1/max(cnt,1). Keeps the expensive IEEE divide
// out of the GEMM inner loop (it is node-dependent and layer-invariant).
// ---------------------------------------------------------------------------
__global__ void invert_cnt(float* __restrict__ cnt, int N) {
    int i = blockIdx.x * blockDim.x + threadIdx.x;
    if (i < N) cnt[i] = 1.0f / fmaxf(cnt[i], 1.0f);
}

// ---------------------------------------------------------------------------
// Feature scatter-add: 32 threads per edge, float4 gather + 4 f32 atomics.
// Feature table (25.6 MB) lives in L2; atomics resolve at L2 atomic units.
// ---------------------------------------------------------------------------
__global__ void scatter_feat(const float* __restrict__ feat,
                             const long long* __restrict__ src,
                             const long long* __restrict__ dst,
                             float* __restrict__ summed, int E) {
    long long gid = (long long)blockIdx.x * blockDim.x + threadIdx.x;
    int e = (int)(gid >> 5);
    if (e >= E) return;
    int c = (int)(gid & 31) * 4;
    long long s = src[e], d = dst[e];
    const float4 v = *(const float4*)(feat + (size_t)s * DFEAT + c);
    float* o = summed + (size_t)d * DFEAT + c;
    atomicAdd(o + 0, v.x);
    atomicAdd(o + 1, v.y);
    atomicAdd(o + 2, v.z);
    atomicAdd(o + 3, v.w);
}

// ---------------------------------------------------------------------------
// out = relu?( (summed * invCnt) @ Wl^T + bl + xin @ Wr^T )
// One wave per 16x16 output tile; 8 waves/block cover all 128 output cols.
// Weight fragments for the wave's column tile are register-resident across
// the persistent row-tile loop (__launch_bounds__(256,1) lifts the VGPR cap
// so the 128 weight VGPRs do not spill; 8 waves/block = 2 waves/SIMD).
// f32 WMMA (16x16x4) keeps exact fp32 math; GEMM is not the bottleneck.
//
// Fragment layouts (ISA 7.12.2, wave32):
//   A (16x4):  lane l -> row m=l%16, holds K = 2*(l/16)+{0,1}   (float2)
//   B (4x16):  lane l -> col n=l%16, holds K = 2*(l/16)+{0,1}   (float2)
//   D (16x16): lane l, vgpr r -> row r+8*(l/16), col l%16
// ---------------------------------------------------------------------------
__global__ void __launch_bounds__(256, 1)
sage_linear(const float* __restrict__ summed,
            const float* __restrict__ invCnt,
            const float* __restrict__ xin,
            const float* __restrict__ Wl,
            const float* __restrict__ bl,
            const float* __restrict__ Wr,
            float* __restrict__ out,
            int nTiles, int doRelu) {
    const int lane  = threadIdx.x & 31;
    const int wave  = threadIdx.x >> 5;   // 0..7 -> output column tile
    const int n0    = wave * 16;
    const int nIdx  = lane & 15;          // B column / A row index within tile
    const int kHalf = (lane >> 4) * 2;    // K sub-offset held by this lane

    // Preload this wave's weight fragments (out = in @ W^T  =>  B[k][n] = W[n][k]).
    v2f wl[32], wr[32];
    const float* wlRow = Wl + (size_t)(n0 + nIdx) * DFEAT + kHalf;
    const float* wrRow = Wr + (size_t)(n0 + nIdx) * DFEAT + kHalf;
#pragma unroll
    for (int k = 0; k < 32; ++k) {
        wl[k] = *(const v2f*)(wlRow + 4 * k);
        wr[k] = *(const v2f*)(wrRow + 4 * k);
    }
    const float bias = bl[n0 + nIdx];

    for (int t = blockIdx.x; t < nTiles; t += gridDim.x) {
        const int row0 = t * 16;
        const int m = row0 + nIdx;        // row this lane sources for A frags
        const float inv = invCnt[m];      // precomputed 1/max(deg,1)
        const float* aggRow = summed + (size_t)m * DFEAT + kHalf;
        const float* xRow   = xin    + (size_t)m * DFEAT + kHalf;

        v8f acc = {};
#pragma unroll
        for (int k = 0; k < 32; ++k) {
            v2f a = *(const v2f*)(aggRow + 4 * k);
            a *= inv;                     // fused mean normalization (v_pk_mul_f32)
            acc = __builtin_amdgcn_wmma_f32_16x16x4_f32(
                false, a, false, wl[k], (short)0, acc, false, false);
            v2f b = *(const v2f*)(xRow + 4 * k);
            acc = __builtin_amdgcn_wmma_f32_16x16x4_f32(
                false, b, false, wr[k], (short)0, acc, false, false);
        }

        const int mBase = row0 + 8 * (lane >> 4);
#pragma unroll
        for (int r = 0; r < 8; ++r) {
            float v = acc[r] + bias;
            if (doRelu) v = fmaxf(v, 0.0f);
            out[(size_t)(mBase + r) * DFEAT + n0 + nIdx] = v;
        }
    }
}

// ---------------------------------------------------------------------------
extern "C" void kernel_launch(void* const* d_in, const int* in_sizes, int n_in,
                              void* d_out, int out_size, void* d_ws, size_t ws_size,
                              hipStream_t stream) {
    (void)n_in; (void)out_size; (void)ws_size;

    const float*     x   = (const float*)d_in[0];
    const long long* ei  = (const long long*)d_in[1];
    const float*     W1l = (const float*)d_in[2];
    const float*     b1  = (const float*)d_in[3];
    const float*     W1r = (const float*)d_in[4];
    const float*     W2l = (const float*)d_in[5];
    const float*     b2  = (const float*)d_in[6];
    const float*     W2r = (const float*)d_in[7];
    const float*     W3l = (const float*)d_in[8];
    const float*     b3  = (const float*)d_in[9];
    const float*     W3r = (const float*)d_in[10];

    const int N = in_sizes[0] / DFEAT;     // 50000
    const int E = in_sizes[1] / 2;         // 800000
    const long long* src = ei;
    const long long* dst = ei + E;

    float* summed = (float*)d_ws;
    float* cnt    = summed + (size_t)N * DFEAT;   // becomes invCnt after invert_cnt
    float* h1     = cnt + N;
    float* h2     = h1 + (size_t)N * DFEAT;
    float* outp   = (float*)d_out;

    const int nTiles    = N / 16;          // 3125 (exact)
    const int linBlocks = 625;             // 5 row-tiles per block, weights reused
    const int scatBlk   = 256;
    const long long scatThreads = (long long)E * 32;
    const int scatGrid  = (int)((scatThreads + scatBlk - 1) / scatBlk);
    const size_t featBytes = (size_t)N * DFEAT * sizeof(float);

    // ---- Degree counts (layer-invariant), then convert to reciprocals ----
    hipMemsetAsync(cnt, 0, (size_t)N * sizeof(float), stream);
    count_deg<<<(E + 255) / 256, 256, 0, stream>>>(dst, cnt, E);
    invert_cnt<<<(N + 255) / 256, 256, 0, stream>>>(cnt, N);

    // ---- Layer 1 ----
    hipMemsetAsync(summed, 0, featBytes, stream);
    scatter_feat<<<scatGrid, scatBlk, 0, stream>>>(x, src, dst, summed, E);
    sage_linear<<<linBlocks, 256, 0, stream>>>(summed, cnt, x, W1l, b1, W1r, h1, nTiles, 1);

    // ---- Layer 2 ----
    hipMemsetAsync(summed, 0, featBytes, stream);
    scatter_feat<<<scatGrid, scatBlk, 0, stream>>>(h1, src, dst, summed, E);
    sage_linear<<<linBlocks, 256, 0, stream>>>(summed, cnt, h1, W2l, b2, W2r, h2, nTiles, 1);

    // ---- Layer 3 (no ReLU, straight to d_out) ----
    hipMemsetAsync(summed, 0, featBytes, stream);
    scatter_feat<<<scatGrid, scatBlk, 0, stream>>>(h2, src, dst, summed, E);
    sage_linear<<<linBlocks, 256, 0, stream>>>(summed, cnt, h2, W3l, b3, W3r, outp, nTiles, 0);
}